// SSIMLoss_137438953922
// MI455X (gfx1250) — compile-verified
//
#include <hip/hip_runtime.h>
#include <hip/hip_bf16.h>

typedef float v2f __attribute__((ext_vector_type(2)));
typedef float v8f __attribute__((ext_vector_type(8)));

#define TILE      64
#define HALO      5
#define IN_W      74          // TILE + 2*HALO live rows
#define IN_STRIDE 76          // padded, 16B-aligned rows
#define HB_STRIDE 68          // hbuf row stride (8-row offset => +32 banks, store conflict-free)
#define HB_ROWS   80          // 5 row-blocks of 16 (rows 74..79 only touched by zero taps)
#define IMG_DIM   512
#define N_PIX     12582912.0f // 16*3*512*512

// normalized 11-tap gaussian (sigma = 1.5); zero outside [0,10]
__device__ __forceinline__ float gtap(int d) {
    float v = 0.0f;
    v = (d == 0 || d == 10) ? 0.00102838f : v;
    v = (d == 1 || d == 9)  ? 0.00759876f : v;
    v = (d == 2 || d == 8)  ? 0.03600077f : v;
    v = (d == 3 || d == 7)  ? 0.10936069f : v;
    v = (d == 4 || d == 6)  ? 0.21300553f : v;
    v = (d == 5)            ? 0.26601172f : v;
    return v;
}

__global__ __launch_bounds__(256) void ssim_main(const float* __restrict__ img1,
                                                 const float* __restrict__ img2,
                                                 float* __restrict__ ws,
                                                 int atomicMode) {
    __shared__ __align__(16) float aT[IN_W * IN_STRIDE];
    __shared__ __align__(16) float bT[IN_W * IN_STRIDE];
    __shared__ __align__(16) float hb[5 * HB_ROWS * HB_STRIDE];
    __shared__ __align__(16) float gtab[7 * 64];   // [chunk][lane] -> v2f taps
    __shared__ float red[8];

    const int tid  = threadIdx.x;
    const int lane = tid & 31;
    const int wid  = tid >> 5;
    const int hs   = (lane >> 4) & 1;   // half-wave select
    const int nn   = lane & 15;

    const int bid   = blockIdx.x;
    const int plane = bid >> 6;                 // 48 planes of 512x512
    const int t64   = bid & 63;
    const int by    = (t64 >> 3) * TILE;
    const int bx    = (t64 & 7) * TILE;
    const size_t pOff = (size_t)plane * (IMG_DIM * IMG_DIM);

    // ---- cooperative halo load (zero padding outside the image) ----
    for (int i = tid; i < IN_W * IN_STRIDE; i += 256) {
        int r  = i / IN_STRIDE;
        int c  = i - r * IN_STRIDE;
        int gy = by + r - HALO;
        int gx = bx + c - HALO;
        float a = 0.0f, b = 0.0f;
        if (gy >= 0 && gy < IMG_DIM && gx >= 0 && gx < IMG_DIM) {
            size_t off = pOff + (size_t)gy * IMG_DIM + gx;
            a = img1[off];
            b = img2[off];
        }
        aT[i] = a;
        bT[i] = b;
    }

    // ---- Toeplitz tap table: gtab[c][lane] = { g(k0-n), g(k0+1-n) } ----
    // f32 A layout: lane holds (M=n, K={4c+2hs, 4c+2hs+1}); B layout mirrored,
    // so the same per-lane values serve both matrix orientations.
    if (tid < 7 * 32) {
        int c7 = tid >> 5;
        int l  = tid & 31;
        int k0 = 4 * c7 + ((l >> 4) & 1) * 2;
        int n  = l & 15;
        gtab[tid * 2]     = gtap(k0 - n);
        gtab[tid * 2 + 1] = gtap(k0 + 1 - n);
    }
    __syncthreads();

    // ---- horizontal 11-tap pass as WMMA: h(16x16) = p(16x28) x Wh(28x16) ----
    // 20 tiles (5 row-blocks x 4 col-blocks), a/b chunks loaded once, 5 maps each.
    for (int t = wid; t < 20; t += 8) {
        int rowbase = (t >> 2) << 4;
        int colbase = (t & 3) << 4;
        int ar = rowbase + nn;
        if (ar > IN_W - 1) ar = IN_W - 1;   // clamp pad rows (hit only by zero taps)
        v2f av[7], bv[7];
#pragma unroll
        for (int c7 = 0; c7 < 7; ++c7) {
            int off = ar * IN_STRIDE + colbase + 4 * c7 + 2 * hs;
            av[c7] = *reinterpret_cast<const v2f*>(&aT[off]);
            bv[c7] = *reinterpret_cast<const v2f*>(&bT[off]);
        }
#pragma unroll
        for (int m = 0; m < 5; ++m) {
            v8f d = {};
#pragma unroll
            for (int c7 = 0; c7 < 7; ++c7) {
                v2f pr = (m == 0) ? av[c7]
                       : (m == 1) ? bv[c7]
                       : (m == 2) ? av[c7] * av[c7]
                       : (m == 3) ? bv[c7] * bv[c7]
                                  : av[c7] * bv[c7];
                v2f tap = *reinterpret_cast<const v2f*>(&gtab[c7 * 64 + lane * 2]);
                d = __builtin_amdgcn_wmma_f32_16x16x4_f32(
                    false, pr, false, tap, (short)0, d, false, false);
            }
#pragma unroll
            for (int e = 0; e < 8; ++e) {
                int orow = rowbase + e + 8 * hs;   // D layout: row = e + 8*halfsel
                hb[(m * HB_ROWS + orow) * HB_STRIDE + colbase + nn] = d[e];
            }
        }
    }
    __syncthreads();

    // ---- vertical 11-tap pass as WMMA: out(16x16) = Wv(16x28) x h(28x16) ----
    v8f acc[5][2] = {};
#pragma unroll
    for (int m = 0; m < 5; ++m) {
        const float* hbm = &hb[m * HB_ROWS * HB_STRIDE];
#pragma unroll
        for (int s = 0; s < 2; ++s) {
            int st  = wid * 2 + s;        // 16 subtiles, 2 per wave
            int ty  = (st >> 2) << 4;
            int cxx = (st & 3) << 4;
#pragma unroll
            for (int c7 = 0; c7 < 7; ++c7) {
                int jb = ty + 4 * c7 + 2 * hs;
                v2f tap = *reinterpret_cast<const v2f*>(&gtab[c7 * 64 + lane * 2]);
                v2f bf;
                bf.x = hbm[jb * HB_STRIDE + cxx + nn];
                bf.y = hbm[(jb + 1) * HB_STRIDE + cxx + nn];
                acc[m][s] = __builtin_amdgcn_wmma_f32_16x16x4_f32(
                    false, tap, false, bf, (short)0, acc[m][s], false, false);
            }
        }
    }

    // ---- elementwise SSIM on identical WMMA layouts + reduction ----
    float part = 0.0f;
#pragma unroll
    for (int s = 0; s < 2; ++s) {
#pragma unroll
        for (int e = 0; e < 8; ++e) {
            float mu1 = acc[0][s][e], mu2 = acc[1][s][e];
            float e11 = acc[2][s][e], e22 = acc[3][s][e], e12 = acc[4][s][e];
            float m11 = mu1 * mu1, m22 = mu2 * mu2, m12 = mu1 * mu2;
            float s1 = e11 - m11, s2 = e22 - m22, s12 = e12 - m12;
            const float C1 = 1e-4f, C2 = 9e-4f;
            float num = (2.0f * m12 + C1) * (2.0f * s12 + C2);
            float den = (m11 + m22 + C1) * (s1 + s2 + C2);
            part += num / den;
        }
    }
    for (int off = 16; off > 0; off >>= 1) part += __shfl_down(part, off, 32);
    if (lane == 0) red[wid] = part;
    __syncthreads();
    if (wid == 0) {
        float v = (lane < 8) ? red[lane] : 0.0f;
        for (int off = 4; off > 0; off >>= 1) v += __shfl_down(v, off, 32);
        if (lane == 0) {
            if (atomicMode) atomicAdd(ws, v);
            else            ws[bid] = v;
        }
    }
}

__global__ void ssim_zero(float* ws) {
    if (threadIdx.x == 0 && blockIdx.x == 0) ws[0] = 0.0f;
}

__global__ __launch_bounds__(256) void ssim_finalize(const float* __restrict__ ws,
                                                     float* __restrict__ out, int n) {
    __shared__ float red[8];
    const int tid = threadIdx.x, lane = tid & 31, wid = tid >> 5;
    float s = 0.0f;
    for (int i = tid; i < n; i += 256) s += ws[i];
    for (int off = 16; off > 0; off >>= 1) s += __shfl_down(s, off, 32);
    if (lane == 0) red[wid] = s;
    __syncthreads();
    if (wid == 0) {
        float v = (lane < 8) ? red[lane] : 0.0f;
        for (int off = 4; off > 0; off >>= 1) v += __shfl_down(v, off, 32);
        if (lane == 0) out[0] = 1.0f - v * (1.0f / N_PIX);
    }
}

extern "C" void kernel_launch(void* const* d_in, const int* in_sizes, int n_in,
                              void* d_out, int out_size, void* d_ws, size_t ws_size,
                              hipStream_t stream) {
    const float* img1 = (const float*)d_in[0];
    const float* img2 = (const float*)d_in[1];
    float* ws  = (float*)d_ws;
    float* out = (float*)d_out;

    const int nBlocks = 48 * 64;  // 3072
    const bool partials = ws_size >= (size_t)nBlocks * sizeof(float);

    if (!partials) ssim_zero<<<1, 32, 0, stream>>>(ws);
    ssim_main<<<nBlocks, 256, 0, stream>>>(img1, img2, ws, partials ? 0 : 1);
    ssim_finalize<<<1, 256, 0, stream>>>(ws, out, partials ? nBlocks : 1);
}